// SinkhornPrior_66614942761141
// MI455X (gfx1250) — compile-verified
//
#include <hip/hip_runtime.h>
#include <math.h>

typedef float v2f __attribute__((ext_vector_type(2)));
typedef float v8f __attribute__((ext_vector_type(8)));

#define N_PTS 4096
#define DIMS  108
#define LOGB  (-8.31776616671934f)   // -log(4096) == loga == logb (uniform weights)
#define EPS_END 1e-4f                // blur^P = 0.01^2

// ---- single-exp online LSE helpers (branchless: exactly one v_exp_f32) ----
__device__ __forceinline__ void lse_add(float& m, float& s, float v) {
  float d  = v - m;                       // m starts at -inf: d=+inf, e=0 -> s=1, m=v
  float e  = __expf(-fabsf(d));
  bool  up = d > 0.0f;
  s = up ? (s * e + 1.0f) : (s + e);
  m = up ? v : m;
}
__device__ __forceinline__ void lse_merge(float& m, float& s, float m2, float s2) {
  float d  = m2 - m;
  float e  = __expf(-fabsf(d));
  bool  up = d > 0.0f;
  s = up ? (s * e + s2) : (s + s2 * e);
  m = up ? m2 : m;
}

// -------------------- squared norms + dual init --------------------
__global__ void sq_init_kernel(const float* __restrict__ X, const float* __restrict__ Y,
                               float* __restrict__ sqx, float* __restrict__ sqy,
                               float* __restrict__ fA, float* __restrict__ gA) {
  int i = blockIdx.x * blockDim.x + threadIdx.x;
  if (i < N_PTS) {
    float sx = 0.f, sy = 0.f;
    for (int d = 0; d < DIMS; ++d) {
      float a = X[(size_t)i * DIMS + d]; sx += a * a;
      float b = Y[(size_t)i * DIMS + d]; sy += b * b;
    }
    sqx[i] = sx; sqy[i] = sy; fA[i] = 0.f; gA[i] = 0.f;
  }
}

// -------------------- bounding-box diameter^2 ----------------------
__global__ void diameter_kernel(const float* __restrict__ X, const float* __restrict__ Y,
                                float* __restrict__ diam2) {
  __shared__ float sd[256];
  int t = threadIdx.x;
  float d2 = 0.f;
  if (t < DIMS) {
    float mn = __builtin_inff(), mx = -__builtin_inff();
    for (int n = 0; n < N_PTS; ++n) { float v = X[(size_t)n * DIMS + t]; mn = fminf(mn, v); mx = fmaxf(mx, v); }
    for (int n = 0; n < N_PTS; ++n) { float v = Y[(size_t)n * DIMS + t]; mn = fminf(mn, v); mx = fmaxf(mx, v); }
    float d = mx - mn; d2 = d * d;
  }
  sd[t] = d2; __syncthreads();
  for (int w = 128; w > 0; w >>= 1) { if (t < w) sd[t] += sd[t + w]; __syncthreads(); }
  if (t == 0) diam2[0] = sd[0];   // eps0 = diameter^2
}

// -------------------- cost matrix via fp32 WMMA --------------------
// One wave computes a 16x16 tile of C = relu(0.5*sqx_i + 0.5*sqy_j - x_i.y_j)
// using v_wmma_f32_16x16x4_f32 accumulation over K=108 in chunks of 4.
__global__ void cost_wmma_kernel(const float* __restrict__ X, const float* __restrict__ Y,
                                 const float* __restrict__ sqx, const float* __restrict__ sqy,
                                 float* __restrict__ C) {
  const int wave = threadIdx.x >> 5;
  const int lane = threadIdx.x & 31;
  const int tile = blockIdx.x * 8 + wave;       // 256x256 tiles total
  const int i0 = (tile >> 8) * 16;
  const int j0 = (tile & 255) * 16;
  const int half = lane >> 4;                   // K-half select (ISA A/B layout)
  const int l    = lane & 15;                   // M (for A) / N (for B)

  const float* xrow = X + (size_t)(i0 + l) * DIMS;
  const float* yrow = Y + (size_t)(j0 + l) * DIMS;

  v8f acc = {};
  for (int k = 0; k < DIMS; k += 4) {
    v2f a, b;
    // A (16x4): VGPR0 holds K = k+2*half, VGPR1 holds K = k+2*half+1
    a.x = xrow[k + 2 * half + 0];
    a.y = xrow[k + 2 * half + 1];
    // B (4x16) = y^T tile, same lane pattern with N on lanes
    b.x = yrow[k + 2 * half + 0];
    b.y = yrow[k + 2 * half + 1];
    acc = __builtin_amdgcn_wmma_f32_16x16x4_f32(
        /*neg_a=*/false, a, /*neg_b=*/false, b,
        /*c_mod=*/(short)0, acc, /*reuse_a=*/false, /*reuse_b=*/false);
  }

  // D layout: VGPR r, lanes 0-15 -> (M=r, N=l); lanes 16-31 -> (M=r+8, N=l)
  const int col  = j0 + l;
  const float hy = 0.5f * sqy[col];
#pragma unroll
  for (int r = 0; r < 8; ++r) {
    int row = i0 + 8 * half + r;
    float v = fmaxf(0.5f * sqx[row] + hy - acc[r], 0.0f);
    C[(size_t)row * N_PTS + col] = v;
  }
}

// -------------------- softmin over rows: f_i ----------------------
// f~_i = -eps*(logb + LSE_j((g_j - C_ij)/eps)); f_new = avg ? 0.5*(f_old+f~) : f~
// One block per row; float4 (b128) loads; single-exp online LSE.
__global__ void softmin_rows_kernel(const float* __restrict__ C, const float* __restrict__ g,
                                    const float* __restrict__ f_old, float* __restrict__ f_new,
                                    const float* __restrict__ diam2, float ratio_k, int avg) {
  __shared__ float sm[256], ss[256];
  const int i = blockIdx.x;
  const int t = threadIdx.x;
  const float eps = fmaxf(diam2[0] * ratio_k, EPS_END);
  const float inv_eps = 1.0f / eps;
  const float* row = C + (size_t)i * N_PTS;

  float m = -__builtin_inff(), s = 0.0f;
  for (int j = t * 4; j < N_PTS; j += 1024) {
    float4 c4 = *(const float4*)(row + j);
    float4 g4 = *(const float4*)(g + j);
    lse_add(m, s, (g4.x - c4.x) * inv_eps);
    lse_add(m, s, (g4.y - c4.y) * inv_eps);
    lse_add(m, s, (g4.z - c4.z) * inv_eps);
    lse_add(m, s, (g4.w - c4.w) * inv_eps);
  }
  sm[t] = m; ss[t] = s; __syncthreads();
  for (int w = 128; w > 0; w >>= 1) {
    if (t < w) {
      float mm = sm[t], sl = ss[t];
      lse_merge(mm, sl, sm[t + w], ss[t + w]);
      sm[t] = mm; ss[t] = sl;
    }
    __syncthreads();
  }
  if (t == 0) {
    float ft = -eps * (LOGB + sm[0] + __logf(ss[0]));
    f_new[i] = avg ? 0.5f * (f_old[i] + ft) : ft;
  }
}

// -------------------- softmin over cols: split-K partials ----------
// Grid (32 col-blocks, 8 row-chunks). Block: 8 stripes x 32 lanes, each
// thread owns 4 adjacent columns (float4 row segments, 512B/wave/iter).
// Writes per-chunk (m, s) partials for each column.
__global__ void softmin_cols_part(const float* __restrict__ C, const float* __restrict__ f,
                                  const float* __restrict__ diam2, float ratio_k,
                                  float* __restrict__ pm, float* __restrict__ ps) {
  __shared__ float smem_m[1024], smem_s[1024];   // [stripe][lane*4+c]
  const int t = threadIdx.x;
  const int lane = t & 31, stripe = t >> 5;
  const int j0 = blockIdx.x * 128 + lane * 4;
  const int chunk = blockIdx.y;                  // 8 chunks of 512 rows
  const float eps = fmaxf(diam2[0] * ratio_k, EPS_END);
  const float inv_eps = 1.0f / eps;

  float m0 = -__builtin_inff(), m1 = m0, m2 = m0, m3 = m0;
  float s0 = 0.f, s1 = 0.f, s2 = 0.f, s3 = 0.f;
  const int ibeg = chunk * 512 + stripe;
  const int iend = chunk * 512 + 512;
  for (int i = ibeg; i < iend; i += 8) {
    float4 c4 = *(const float4*)(C + (size_t)i * N_PTS + j0);
    float fi = f[i];
    lse_add(m0, s0, (fi - c4.x) * inv_eps);
    lse_add(m1, s1, (fi - c4.y) * inv_eps);
    lse_add(m2, s2, (fi - c4.z) * inv_eps);
    lse_add(m3, s3, (fi - c4.w) * inv_eps);
  }
  const int base = stripe * 128 + lane * 4;
  smem_m[base + 0] = m0; smem_s[base + 0] = s0;
  smem_m[base + 1] = m1; smem_s[base + 1] = s1;
  smem_m[base + 2] = m2; smem_s[base + 2] = s2;
  smem_m[base + 3] = m3; smem_s[base + 3] = s3;
  __syncthreads();
  if (stripe == 0) {                              // t < 32: merge 8 stripes
    float rm[4], rs[4];
#pragma unroll
    for (int c = 0; c < 4; ++c) { rm[c] = smem_m[lane * 4 + c]; rs[c] = smem_s[lane * 4 + c]; }
    for (int q = 1; q < 8; ++q) {
#pragma unroll
      for (int c = 0; c < 4; ++c)
        lse_merge(rm[c], rs[c], smem_m[q * 128 + lane * 4 + c], smem_s[q * 128 + lane * 4 + c]);
    }
    float4 vm = make_float4(rm[0], rm[1], rm[2], rm[3]);
    float4 vs = make_float4(rs[0], rs[1], rs[2], rs[3]);
    *(float4*)(pm + (size_t)chunk * N_PTS + j0) = vm;
    *(float4*)(ps + (size_t)chunk * N_PTS + j0) = vs;
  }
}

// Merge the 8 row-chunk partials per column and finish the softmin.
__global__ void softmin_cols_combine(const float* __restrict__ pm, const float* __restrict__ ps,
                                     const float* __restrict__ g_old, float* __restrict__ g_new,
                                     const float* __restrict__ diam2, float ratio_k, int avg) {
  const int j = blockIdx.x * 256 + threadIdx.x;
  const float eps = fmaxf(diam2[0] * ratio_k, EPS_END);
  float m = pm[j], s = ps[j];
  for (int q = 1; q < 8; ++q)
    lse_merge(m, s, pm[(size_t)q * N_PTS + j], ps[(size_t)q * N_PTS + j]);
  float gt = -eps * (LOGB + m + __logf(s));
  g_new[j] = avg ? 0.5f * (g_old[j] + gt) : gt;
}

// -------------------- final scalar: mean(f)+mean(g) ----------------
__global__ void mean_kernel(const float* __restrict__ f, const float* __restrict__ g,
                            float* __restrict__ out) {
  __shared__ float sb[256];
  int t = threadIdx.x;
  float s = 0.f;
  for (int i = t; i < N_PTS; i += 256) s += f[i] + g[i];
  sb[t] = s; __syncthreads();
  for (int w = 128; w > 0; w >>= 1) { if (t < w) sb[t] += sb[t + w]; __syncthreads(); }
  if (t == 0) out[0] = sb[0] / (float)N_PTS;   // N == M -> (sum f + sum g)/N
}

// ------------------------------------------------------------------
extern "C" void kernel_launch(void* const* d_in, const int* in_sizes, int n_in,
                              void* d_out, int out_size, void* d_ws, size_t ws_size,
                              hipStream_t stream) {
  const float* X = (const float*)d_in[0];   // fake_data     [4096,108]
  const float* Y = (const float*)d_in[1];   // reference_img [4096,108]
  float* out = (float*)d_out;

  float* ws = (float*)d_ws;
  const size_t NC = (size_t)N_PTS * N_PTS;
  float* C     = ws;                 // 64 MB, L2-resident
  float* sqx   = ws + NC;
  float* sqy   = sqx + N_PTS;
  float* fA    = sqy + N_PTS;
  float* gA    = fA + N_PTS;
  float* fB    = gA + N_PTS;
  float* gB    = fB + N_PTS;
  float* diam2 = gB + N_PTS;
  float* pm    = diam2 + 16;         // [8][4096] column partial maxes
  float* ps    = pm + 8 * N_PTS;     // [8][4096] column partial sums

  sq_init_kernel<<<16, 256, 0, stream>>>(X, Y, sqx, sqy, fA, gA);
  diameter_kernel<<<1, 256, 0, stream>>>(X, Y, diam2);
  cost_wmma_kernel<<<8192, 256, 0, stream>>>(X, Y, sqx, sqy, C);

  // 24 symmetric (averaged) steps, eps_k = max(diam2 * 0.25^k, EPS_END)
  float* fc = fA; float* gc = gA; float* fn = fB; float* gn = gB;
  float rk = 1.0f;
  for (int k = 0; k < 24; ++k) {
    softmin_rows_kernel<<<N_PTS, 256, 0, stream>>>(C, gc, fc, fn, diam2, rk, 1);
    softmin_cols_part<<<dim3(32, 8), 256, 0, stream>>>(C, fc, diam2, rk, pm, ps);
    softmin_cols_combine<<<16, 256, 0, stream>>>(pm, ps, gc, gn, diam2, rk, 1);
    float* tf = fc; fc = fn; fn = tf;
    float* tg = gc; gc = gn; gn = tg;
    rk *= 0.25f;
  }

  // final non-averaged updates at eps_end (ratio 0 -> eps clamps to EPS_END);
  // final g uses the NEW f, matching the reference.
  softmin_rows_kernel<<<N_PTS, 256, 0, stream>>>(C, gc, fc, fn, diam2, 0.0f, 0);
  softmin_cols_part<<<dim3(32, 8), 256, 0, stream>>>(C, fn, diam2, 0.0f, pm, ps);
  softmin_cols_combine<<<16, 256, 0, stream>>>(pm, ps, gc, gn, diam2, 0.0f, 0);

  mean_kernel<<<1, 256, 0, stream>>>(fn, gn, out);
}